// BP_Decoder_53961969107423
// MI455X (gfx1250) — compile-verified
//
#include <hip/hip_runtime.h>
#include <stdint.h>

// Problem constants (match reference)
#define KN_VN   1024
#define KM_CN   512
#define KE      4096
#define KDC     8
#define KDV     4
#define KITER   5

// Tiling
#define KTB     8            // batch columns per block (lanes b = tid & 7)
#define KMST    9            // msg LDS stride (floats per edge) -> conflict-free CN phase
#define KTHREADS 256
#define LLRT_STRIDE 1028     // 1024 + 4 TDM pad dwords per batch row

typedef __attribute__((ext_vector_type(4))) unsigned int v4u;
typedef __attribute__((ext_vector_type(8))) int          v8i;
typedef __attribute__((ext_vector_type(4))) int          v4i;

__device__ __forceinline__ float my_tanh(float x) {
#if __has_builtin(__builtin_amdgcn_tanhf)
  return __builtin_amdgcn_tanhf(x);     // v_tanh_f32 (gfx1250 TRANS op)
#else
  return tanhf(x);
#endif
}

// atanh2(x) = log(clip(1+x)/clip(1-x)) = ln2 * (log2(e1) - log2(e2))
__device__ __forceinline__ float atanh2f(float x) {
  float e1 = fminf(fmaxf(1.0f + x, 1e-7f), 2.0f - 1e-7f);
  float e2 = fminf(fmaxf(1.0f - x, 1e-7f), 2.0f - 1e-7f);
  return 0.69314718055994531f * (__log2f(e1) - __log2f(e2));
}

// TDM: async-load an 8x1024 f32 tile (row stride 1024 elems) into LDS with
// 1 dword of padding per 256 dwords stored -> LDS addr(b,v) = b*1028 + v + (v>>8)
// (the TDM pad feature performs the bank-conflict-avoiding skew for free).
__device__ __forceinline__ void tdm_issue_llr_tile(const float* gsrc, uint32_t lds_off_bytes) {
  uint64_t ga = (uint64_t)(uintptr_t)gsrc;
  v4u g0;
  g0.x = 1u;                                     // count = 1 (valid descriptor)
  g0.y = lds_off_bytes;                          // lds_addr
  g0.z = (uint32_t)(ga & 0xFFFFFFFFu);           // global_addr[31:0]
  g0.w = (uint32_t)((ga >> 32) & 0x1FFFFFFull)   // global_addr[56:32]
       | (2u << 30);                             // type = 2 ("image")
  v8i g1;
  g1[0] = (2 << 16)        // data_size = 4 bytes
        | (1 << 20)        // pad_enable
        | (7 << 22)        // pad_interval: 256 dwords
        | (0 << 25);       // pad_amount: 1 dword
  g1[1] = (int)(KN_VN << 16);          // tensor_dim0[15:0] = 1024 (bits 79:48)
  g1[2] = (int)(KTB   << 16);          // tensor_dim1[15:0] = 8    (bits 111:80)
  g1[3] = (int)(KN_VN << 16);          // tile_dim0 = 1024         (bits 127:112)
  g1[4] = KTB;                         // tile_dim1 = 8, tile_dim2 = 0
  g1[5] = KN_VN;                       // tensor_dim0_stride = 1024 elems
  g1[6] = 0;
  g1[7] = 0;
  v4i gz = {0, 0, 0, 0};
#if __has_builtin(__builtin_amdgcn_tensor_load_to_lds)
#  if defined(__clang_major__) && __clang_major__ >= 23
  v8i gz8 = {0, 0, 0, 0, 0, 0, 0, 0};
  __builtin_amdgcn_tensor_load_to_lds(g0, g1, gz, gz, gz8, 0);
#  else
  __builtin_amdgcn_tensor_load_to_lds(g0, g1, gz, gz, 0);
#  endif
#else
  asm volatile("tensor_load_to_lds %0, %1, %2, %3"
               :: "s"(g0), "s"(g1), "s"(gz), "s"(gz) : "memory");
#endif
}

extern "C" __global__ __launch_bounds__(KTHREADS)
void ldpc_bp_kernel(const float* __restrict__ llr,
                    const int*   __restrict__ ro_idx,
                    const int*   __restrict__ vn_edges,
                    float*       __restrict__ out)
{
  extern __shared__ char smem_raw[];
  float* msg  = (float*)smem_raw;                                   // KE * KMST floats (144 KB)
  float* llrT = (float*)(smem_raw + (size_t)KE * KMST * 4);         // KTB * 1028 floats (32.9 KB)
  int*   tbl  = (int*)  (smem_raw + (size_t)KE * KMST * 4
                                  + (size_t)KTB * LLRT_STRIDE * 4); // KN_VN int4 (16 KB)

  const int tid = threadIdx.x;
  const int b0  = blockIdx.x * KTB;
  const int bb  = tid & (KTB - 1);     // batch column within tile
  const int vb  = tid >> 3;            // 0..31 stripe base

  // Wave 0 kicks off the async tensor DMA of the llr tile FIRST, so it
  // overlaps the global index gathers below.
  if (tid < 32) {
    tdm_issue_llr_tile(llr + (size_t)b0 * KN_VN, (uint32_t)(uintptr_t)llrT);
  }

  // Build per-VN edge table (4 edges of each variable) while TDM is in flight.
  // Batched: all vn_edges loads in flight, then all ro_idx b128 gathers.
  {
    int e0[KN_VN / KTHREADS];
    #pragma unroll
    for (int k = 0; k < KN_VN / KTHREADS; ++k)
      e0[k] = vn_edges[tid + KTHREADS * k];
    #pragma unroll
    for (int k = 0; k < KN_VN / KTHREADS; ++k)
      ((int4*)tbl)[tid + KTHREADS * k] = ((const int4*)ro_idx)[e0[k]];
  }

  if (tid < 32) {
    __builtin_amdgcn_s_wait_tensorcnt(0);        // s_wait_tensorcnt 0x0
  }
  __syncthreads();

  // ---- init: h_m[e] = tanh(0.5 * llr[b, vn(e)]) (VN-centric, 4 edges share h_r)
  for (int k = 0; k < (KN_VN * KTB) / KTHREADS; ++k) {
    int v = vb + 32 * k;
    float r = llrT[bb * LLRT_STRIDE + v + (v >> 8)];
    int4 e = ((const int4*)tbl)[v];
    float t = my_tanh(0.5f * r);
    msg[e.x * KMST + bb] = t;
    msg[e.y * KMST + bb] = t;
    msg[e.z * KMST + bb] = t;
    msg[e.w * KMST + bb] = t;
  }
  __syncthreads();

  for (int it = 0; it < KITER; ++it) {
    // ---- check-node phase: h_e = atanh2(prod of other 7 h_m), in place.
    // Edges of check c are exactly [8c, 8c+8) (row-major np.where ordering).
    for (int k = 0; k < (KM_CN * KTB) / KTHREADS; ++k) {
      int c = vb + 32 * k;
      int base = c * KDC * KMST + bb;
      float m[KDC], ex[KDC];
      #pragma unroll
      for (int i = 0; i < KDC; ++i) m[i] = msg[base + i * KMST];
      float pre = 1.0f, suf = 1.0f;
      #pragma unroll
      for (int i = 0; i < KDC; ++i) { ex[i] = pre; pre *= m[i]; }
      #pragma unroll
      for (int i = KDC - 1; i >= 0; --i) { ex[i] *= suf; suf *= m[i]; }
      #pragma unroll
      for (int i = 0; i < KDC; ++i) msg[base + i * KMST] = atanh2f(ex[i]);
    }
    __syncthreads();

    if (it + 1 < KITER) {
      // ---- variable-node phase: h_m = tanh(0.5*(sum other 3 h_e + h_r)), in place.
      for (int k = 0; k < (KN_VN * KTB) / KTHREADS; ++k) {
        int v = vb + 32 * k;
        int4 e = ((const int4*)tbl)[v];
        float h0 = msg[e.x * KMST + bb];
        float h1 = msg[e.y * KMST + bb];
        float h2 = msg[e.z * KMST + bb];
        float h3 = msg[e.w * KMST + bb];
        float r  = llrT[bb * LLRT_STRIDE + v + (v >> 8)];
        float S  = ((h0 + h1) + (h2 + h3)) + r;   // includes h_r
        msg[e.x * KMST + bb] = my_tanh(0.5f * (S - h0));
        msg[e.y * KMST + bb] = my_tanh(0.5f * (S - h1));
        msg[e.z * KMST + bb] = my_tanh(0.5f * (S - h2));
        msg[e.w * KMST + bb] = my_tanh(0.5f * (S - h3));
      }
      __syncthreads();
    }
  }

  // ---- output: out[b0+b, v] = sum of 4 h_e of v + llr[b0+b, v] (coalesced stores)
  for (int k = 0; k < (KN_VN * KTB) / KTHREADS; ++k) {
    int i = tid + KTHREADS * k;
    int b = i >> 10;
    int v = i & (KN_VN - 1);
    int4 e = ((const int4*)tbl)[v];
    float S = (msg[e.x * KMST + b] + msg[e.y * KMST + b])
            + (msg[e.z * KMST + b] + msg[e.w * KMST + b]);
    out[(size_t)(b0 + b) * KN_VN + v] = S + llrT[b * LLRT_STRIDE + v + (v >> 8)];
  }
}

extern "C" void kernel_launch(void* const* d_in, const int* in_sizes, int n_in,
                              void* d_out, int out_size, void* d_ws, size_t ws_size,
                              hipStream_t stream) {
  // setup_inputs order: llr, edge_vn, hm_idx, he_idx, ro_idx, vn_edges
  const float* llr      = (const float*)d_in[0];
  const int*   ro_idx   = (const int*)d_in[4];
  const int*   vn_edges = (const int*)d_in[5];
  float*       out      = (float*)d_out;
  (void)d_ws; (void)ws_size; (void)n_in; (void)out_size;

  int batch = in_sizes[0] / KN_VN;                 // 2048
  size_t smem = (size_t)KE * KMST * 4              // msg      147456 B
              + (size_t)KTB * LLRT_STRIDE * 4      // llr tile  32896 B
              + (size_t)KN_VN * 4 * 4;             // edge tbl  16384 B
  dim3 grid(batch / KTB);                          // 256 blocks
  ldpc_bp_kernel<<<grid, KTHREADS, smem, stream>>>(llr, ro_idx, vn_edges, out);
}